// ObjectModule_46136538694238
// MI455X (gfx1250) — compile-verified
//
#include <hip/hip_runtime.h>
#include <stdint.h>

// Problem constants (B=8, K=4, N=131072)
#define BKN      4194304          // B*K*N
#define CHUNK    4096             // elements per block (divides N -> 32 chunks per (b,k))
#define NBLK     (BKN / CHUNK)    // 1024
#define TPB      256              // 8 waves
#define SEG      512              // elements per wave segment (CHUNK / 8)
#define WROUNDS  (SEG / 32)       // 16 rounds of 32 lanes per wave
#define NEG_PAD  (-1000.0f)
#define NEG_IDLE (-100000000.0f)

// ---------------------------------------------------------------------------
// Pass 1: per-chunk popcounts of both masks. sums[0..NBLK-1] = surface,
// sums[NBLK..2*NBLK-1] = air.
// ---------------------------------------------------------------------------
__global__ __launch_bounds__(TPB)
void count_kernel(const unsigned char* __restrict__ ms,
                  const unsigned char* __restrict__ ma,
                  unsigned* __restrict__ sums)
{
    const int c = blockIdx.x, t = threadIdx.x;
    const uint4* ps = (const uint4*)(ms + (size_t)c * CHUNK);
    const uint4* pa = (const uint4*)(ma + (size_t)c * CHUNK);
    uint4 xs = ps[t];
    uint4 xa = pa[t];
    const unsigned M = 0x01010101u;   // bool bytes are 0/1 -> popc counts them
    unsigned cs = __popc(xs.x & M) + __popc(xs.y & M) + __popc(xs.z & M) + __popc(xs.w & M);
    unsigned ca = __popc(xa.x & M) + __popc(xa.y & M) + __popc(xa.z & M) + __popc(xa.w & M);

    __shared__ unsigned red[TPB];
    red[t] = cs | (ca << 16);         // each half <= 4096, fits in 16 bits
    __syncthreads();
    for (int off = TPB / 2; off > 0; off >>= 1) {
        if (t < off) red[t] += red[t + off];
        __syncthreads();
    }
    if (t == 0) {
        unsigned v = red[0];
        sums[c]        = v & 0xFFFFu;
        sums[NBLK + c] = v >> 16;
    }
}

// ---------------------------------------------------------------------------
// Pass 2: exclusive scan of the 1024 chunk sums (both masks packed into u64).
// ---------------------------------------------------------------------------
__global__ __launch_bounds__(NBLK)
void scan_kernel(const unsigned* __restrict__ sums,
                 unsigned* __restrict__ offs)
{
    const int t = threadIdx.x;
    __shared__ unsigned long long s[NBLK];
    unsigned long long v = (unsigned long long)sums[t] |
                           ((unsigned long long)sums[NBLK + t] << 32);
    s[t] = v;
    __syncthreads();
    for (int off = 1; off < NBLK; off <<= 1) {
        unsigned long long x = (t >= off) ? s[t - off] : 0ull;
        __syncthreads();
        s[t] += x;                    // per-half totals < 2^32 -> no cross-carry
        __syncthreads();
    }
    unsigned long long e = (t == 0) ? 0ull : s[t - 1];
    offs[t]        = (unsigned)(e & 0xFFFFFFFFull);
    offs[NBLK + t] = (unsigned)(e >> 32);
}

// ---------------------------------------------------------------------------
// Pass 3: expand + gather + write.
//  - masks staged into LDS via CDNA5 async global->LDS DMA (ASYNCcnt path)
//  - each wave owns a contiguous 512-element segment: after a one-time
//    8-wave exclusive scan, the 16-round hot loop is barrier-free (wave32
//    ballot prefix only)
//  - outputs written with non-temporal TH hints (write-once stream)
// ---------------------------------------------------------------------------
__global__ __launch_bounds__(TPB)
void emit_kernel(const float* __restrict__ rgb,
                 const float* __restrict__ ls_in,
                 const float* __restrict__ la_in,
                 const float* __restrict__ idle_states,
                 const unsigned char* __restrict__ ms,
                 const unsigned char* __restrict__ ma,
                 const unsigned* __restrict__ offs,
                 float* __restrict__ out)
{
    const int c = blockIdx.x, t = threadIdx.x;
    const int lane = t & 31;
    const int wave = t >> 5;

    __shared__ __align__(16) unsigned char lmask[2 * CHUNK];  // 8 KB of 320 KB/WGP
    __shared__ unsigned red[TPB];
    __shared__ unsigned wtot[8];

    // --- async-DMA the two 4KB mask slices into LDS (16B per lane per op) ---
    {
        unsigned long long gs = (unsigned long long)(uintptr_t)(ms + (size_t)c * CHUNK + t * 16);
        unsigned long long ga = (unsigned long long)(uintptr_t)(ma + (size_t)c * CHUNK + t * 16);
        // low 32 bits of a generic LDS pointer == wave-relative LDS byte offset
        unsigned ldss = (unsigned)(uintptr_t)&lmask[t * 16];
        unsigned ldsa = (unsigned)(uintptr_t)&lmask[CHUNK + t * 16];
        asm volatile("global_load_async_to_lds_b128 %0, %1, off"
                     :: "v"(ldss), "v"(gs) : "memory");
        asm volatile("global_load_async_to_lds_b128 %0, %1, off"
                     :: "v"(ldsa), "v"(ga) : "memory");
#if __has_builtin(__builtin_amdgcn_s_wait_asynccnt)
        __builtin_amdgcn_s_wait_asynccnt(0);
#else
        asm volatile("s_wait_asynccnt 0x0" ::: "memory");
#endif
    }
    __syncthreads();

    // --- one-time per-wave segment counts (lane l counts 16B of its wave's
    //     512B segment) and exclusive scan across the 8 waves ---
    {
        const uint4 xs = *(const uint4*)&lmask[wave * SEG + lane * 16];
        const uint4 xa = *(const uint4*)&lmask[CHUNK + wave * SEG + lane * 16];
        const unsigned M = 0x01010101u;
        unsigned cs = __popc(xs.x & M) + __popc(xs.y & M) + __popc(xs.z & M) + __popc(xs.w & M);
        unsigned ca = __popc(xa.x & M) + __popc(xa.y & M) + __popc(xa.z & M) + __popc(xa.w & M);
        red[t] = cs | (ca << 16);     // wave total <= 512 per half: fits 16 bits
    }
    __syncthreads();
    if (t < 8) {
        unsigned v = 0;
        #pragma unroll
        for (int i = 0; i < 32; ++i) v += red[t * 32 + i];
        wtot[t] = v;
    }
    __syncthreads();

    unsigned run_s, run_a;
    {
        unsigned woff = 0;
        #pragma unroll
        for (int w = 0; w < 8; ++w)
            if (w < wave) woff += wtot[w];
        run_s = offs[c]        + (woff & 0xFFFFu);
        run_a = offs[NBLK + c] + (woff >> 16);
    }

    const float idle    = idle_states[c >> 5];   // 32 chunks per (b,k) row
    const float alive   = 1.0f - idle;
    const float lg_idle = idle * NEG_IDLE;

    float* __restrict__ out_rgb = out;
    float* __restrict__ out_ls  = out + (size_t)BKN * 3;
    float* __restrict__ out_la  = out + (size_t)BKN * 4;

    const unsigned ltmask = (1u << lane) - 1u;

    // --- barrier-free hot loop: each wave walks its own 512-element segment ---
    for (int r = 0; r < WROUNDS; ++r) {
        // prefetch the upcoming gather stream (global_prefetch_b8)
        __builtin_prefetch(&rgb[(size_t)run_s * 3], 0, 0);
        __builtin_prefetch(&la_in[run_a], 0, 0);

        const int e = wave * SEG + r * 32 + lane;  // element within chunk
        const size_t idx = (size_t)c * CHUNK + e;  // flat (b,k,n)
        const bool s_on = lmask[e] != 0;
        const bool a_on = lmask[CHUNK + e] != 0;

        const unsigned bal_s = __builtin_amdgcn_ballot_w32(s_on);
        const unsigned bal_a = __builtin_amdgcn_ballot_w32(a_on);
        const unsigned pos_s = run_s + __popc(bal_s & ltmask);
        const unsigned pos_a = run_a + __popc(bal_a & ltmask);

        float r0 = 0.0f, r1 = 0.0f, r2 = 0.0f;
        float lsv = NEG_PAD, lav = NEG_PAD;
        if (s_on) {
            r0  = rgb[(size_t)pos_s * 3 + 0] * alive;
            r1  = rgb[(size_t)pos_s * 3 + 1] * alive;
            r2  = rgb[(size_t)pos_s * 3 + 2] * alive;
            lsv = ls_in[pos_s] * alive + lg_idle;
        }
        if (a_on) {
            lav = la_in[pos_a] * alive + lg_idle;
        }

        // write-once output stream: non-temporal TH hint keeps L2 for the
        // masks / gather sources
        __builtin_nontemporal_store(r0,  &out_rgb[idx * 3 + 0]);
        __builtin_nontemporal_store(r1,  &out_rgb[idx * 3 + 1]);
        __builtin_nontemporal_store(r2,  &out_rgb[idx * 3 + 2]);
        __builtin_nontemporal_store(lsv, &out_ls[idx]);
        __builtin_nontemporal_store(lav, &out_la[idx]);

        run_s += __popc(bal_s);
        run_a += __popc(bal_a);
    }
}

// ---------------------------------------------------------------------------
// Inputs (setup_inputs order): rgb(ts,3) f32, logits_surface(ts,1) f32,
// logits_air(ta,1) f32, idle_states(8,4) f32, mask_eval_surface(8,4,131072)
// bool, mask_eval_air(8,4,131072) bool.
// Output: rgb_out | logits_surface_out | logits_air_out, flat f32.
// Scratch use: 4*NBLK u32 = 16 KB of d_ws.
// ---------------------------------------------------------------------------
extern "C" void kernel_launch(void* const* d_in, const int* in_sizes, int n_in,
                              void* d_out, int out_size, void* d_ws, size_t ws_size,
                              hipStream_t stream)
{
    const float* rgb  = (const float*)d_in[0];
    const float* ls   = (const float*)d_in[1];
    const float* la   = (const float*)d_in[2];
    const float* idle = (const float*)d_in[3];
    const unsigned char* msk_s = (const unsigned char*)d_in[4];  // jax bool = 1 byte
    const unsigned char* msk_a = (const unsigned char*)d_in[5];
    float* out = (float*)d_out;

    unsigned* sums = (unsigned*)d_ws;     // [2*NBLK]
    unsigned* offs = sums + 2 * NBLK;     // [2*NBLK]

    count_kernel<<<NBLK, TPB, 0, stream>>>(msk_s, msk_a, sums);
    scan_kernel<<<1, NBLK, 0, stream>>>(sums, offs);
    emit_kernel<<<NBLK, TPB, 0, stream>>>(rgb, ls, la, idle, msk_s, msk_a, offs, out);
}